// RationalQuadraticSpline_40973988004342
// MI455X (gfx1250) — compile-verified
//
#include <hip/hip_runtime.h>
#include <hip/hip_bf16.h>
#include <cstdint>

// ---------------------------------------------------------------------------
// Rational-quadratic spline (neural spline flow) for MI455X / gfx1250.
//
// Phase 1 (rqs_param_kernel): one wave32 per variable. Softmax (shfl-xor
//   reductions) + inclusive scan (shfl-up) build the knot tables in d_ws,
//   layout [V=64][stride=193] f32 (odd stride -> LDS-bank friendly).
// Phase 2 (rqs_eval_kernel): each block DMAs the whole 48.25KB table into
//   LDS with a single tensor_load_to_lds (TDM), waits on TENSORcnt, then
//   grid-strides over B*V elements with float4 (b128) loads/stores, a
//   5-step binary bin search in LDS and ~25 VALU ops + 2 hw logs/element.
//   Roofline: 16MB in + 32MB out ~= 2.1us at 23.3 TB/s -> bandwidth bound.
// ---------------------------------------------------------------------------

#define KBINS     30
#define NV        64
#define TSTRIDE   193                 // floats per variable row (193 mod 64 == 1)
#define OFF_CUMW  0                   // 31 floats: knots, last = 1 + EPS (search bounds)
#define OFF_INVW  31                  // 30 floats: 1/width
#define OFF_CUMH  61                  // 31 floats: cumheights
#define OFF_DELTA 92                  // 30 floats: height/width
#define OFF_DER   122                 // 31 floats: derivatives (ends pinned to 1)
#define OFF_H     153                 // 30 floats: heights
#define TBL_FLOATS (NV * TSTRIDE)     // 12352 floats = 49408 bytes

typedef uint32_t u32;
typedef u32 u32x4 __attribute__((ext_vector_type(4)));
typedef u32 u32x8 __attribute__((ext_vector_type(8)));

// ---------------------------------------------------------------------------
// Phase 1: wave-parallel softmax + cumsum (lane k <-> bin k)
// ---------------------------------------------------------------------------
// Builds cum[0..30] (cum[30] = lastStore) and returns this lane's bin width.
__device__ __forceinline__ float cum_scan(const float* __restrict__ a,
                                          float* __restrict__ outCum,
                                          int lane, float lastStore) {
    float x = (lane < KBINS) ? a[lane] : -3.4e38f;
    // wave32 max
    float m = x;
    #pragma unroll
    for (int off = 16; off; off >>= 1) m = fmaxf(m, __shfl_xor(m, off));
    float e = (lane < KBINS) ? __expf(x - m) : 0.f;
    // wave32 sum
    float s = e;
    #pragma unroll
    for (int off = 16; off; off >>= 1) s += __shfl_xor(s, off);
    // floored softmax bin size
    float p = (lane < KBINS) ? (1e-3f + (1.f - 1e-3f * (float)KBINS) * (e / s)) : 0.f;
    // inclusive scan (Hillis-Steele over 32 lanes)
    float c = p;
    #pragma unroll
    for (int off = 1; off < 32; off <<= 1) {
        float t = __shfl_up(c, off);
        if (lane >= off) c += t;
    }
    // cp[lane+1] = c, with last knot pinned to hi (and EPS for the bounds copy)
    float cv   = (lane == KBINS - 1) ? 1.f : c;   // pinned knot value (for widths)
    float prev = __shfl_up(c, 1);                  // cp[lane] (raw, never pinned for lane<=29)
    if (lane == 0) { prev = 0.f; outCum[0] = 0.f; }
    if (lane < KBINS) outCum[lane + 1] = (lane == KBINS - 1) ? lastStore : c;
    return cv - prev;                              // width of bin `lane`
}

__global__ __launch_bounds__(32) void rqs_param_kernel(const float* __restrict__ uw,
                                                       const float* __restrict__ uh,
                                                       const float* __restrict__ ud,
                                                       float* __restrict__ ws) {
    const int v    = (int)blockIdx.x;              // one wave32 per variable
    const int lane = (int)threadIdx.x;
    float* row = ws + (size_t)v * TSTRIDE;

    float wd = cum_scan(uw + v * KBINS, row + OFF_CUMW, lane, 1.f + 1e-6f);
    float ht = cum_scan(uh + v * KBINS, row + OFF_CUMH, lane, 1.f);

    if (lane < KBINS) {
        row[OFF_INVW  + lane] = 1.f / wd;
        row[OFF_DELTA + lane] = ht / wd;
        row[OFF_H     + lane] = ht;
    }
    // derivatives: ends pinned so MIN_D + softplus(const) == 1
    if (lane <= KBINS) {
        float dv;
        if (lane == 0 || lane == KBINS) {
            dv = 1.f;
        } else {
            float x = ud[v * (KBINS - 1) + (lane - 1)];
            dv = 1e-3f + ((x > 20.f) ? x : log1pf(__expf(x)));
        }
        row[OFF_DER + lane] = dv;
    }
}

// ---------------------------------------------------------------------------
// Per-element spline evaluation from the LDS table of one variable.
// ---------------------------------------------------------------------------
__device__ __forceinline__ void eval_one(const float* __restrict__ T, float x,
                                         float& o, float& l) {
    // branchless 5-step binary search over 31 bounds T[0..30]
    int k = 0;
    if (T[16]     <= x) k = 16;
    if (T[k + 8]  <= x) k += 8;
    if (T[k + 4]  <= x) k += 4;
    if (T[k + 2]  <= x) k += 2;
    if (k < KBINS && T[k + 1] <= x) k += 1;
    int b = (k > KBINS - 1) ? (KBINS - 1) : k;

    float icw  = T[b];
    float invw = T[OFF_INVW  + b];
    float ich  = T[OFF_CUMH  + b];
    float idl  = T[OFF_DELTA + b];
    float d0   = T[OFF_DER   + b];
    float d1   = T[OFF_DER   + b + 1];
    float ih   = T[OFF_H     + b];

    float th   = (x - icw) * invw;
    float omt  = 1.f - th;
    float t1mt = th * omt;
    float th2  = th * th;

    float num  = ih * fmaf(idl, th2, d0 * t1mt);
    float den  = fmaf(d0 + d1 - 2.f * idl, t1mt, idl);
    float so   = ich + num / den;
    float dn   = (idl * idl) * fmaf(d1, th2, fmaf(2.f * idl, t1mt, d0 * omt * omt));
    float la   = __logf(dn) - 2.f * __logf(den);

    bool inside = (x >= 0.f) && (x <= 1.f);
    o = inside ? so : x;     // outside: slope 1, zero offset -> identity
    l = inside ? la : 0.f;
}

// ---------------------------------------------------------------------------
// Phase 2: main bandwidth-bound kernel
// ---------------------------------------------------------------------------
__global__ __launch_bounds__(256) void rqs_eval_kernel(
    const float* __restrict__ in, const float* __restrict__ ws,
    float* __restrict__ out, float* __restrict__ lad, int n4, int N) {

    __shared__ __align__(16) float S[TBL_FLOATS];

    // --- TDM: one tensor_load_to_lds DMAs the whole 49408B table into LDS ---
    // Issued by wave 0 only (TDM ignores EXEC; one instruction per wave).
    if (threadIdx.x < 32u) {
        u32 ldsAddr = (u32)(size_t)(&S[0]);
        unsigned long long ga = (unsigned long long)(size_t)ws;
        const u32 td0 = (u32)TBL_FLOATS;   // elements (data_size = 4B)

        u32x4 g0;
        g0[0] = 1u;                                   // count=1 valid descriptor
        g0[1] = ldsAddr;                              // lds_addr (bytes)
        g0[2] = (u32)ga;                              // global_addr[31:0]
        g0[3] = (u32)(ga >> 32) | 0x80000000u;        // global_addr[56:32] | type=2

        u32x8 g1;
        g1[0] = 2u << 16;                             // data_size=2 (4B), wg_mask=0
        g1[1] = (td0 & 0xFFFFu) << 16;                // tensor_dim0[15:0]
        g1[2] = ((td0 >> 16) & 0xFFFFu) | (1u << 16); // tensor_dim0[31:16], tensor_dim1=1
        g1[3] = (td0 & 0xFFFFu) << 16;                // tile_dim0 = 12352
        g1[4] = 1u;                                   // tile_dim1=1, tile_dim2=0
        g1[5] = td0;                                  // tensor_dim0_stride[31:0]
        g1[6] = 0u;
        g1[7] = 0u;

        asm volatile("tensor_load_to_lds %0, %1" :: "s"(g0), "s"(g1) : "memory");
        __builtin_amdgcn_s_wait_tensorcnt(0);
    }
    __syncthreads();

    const int gsz = (int)(gridDim.x * blockDim.x);

    // float4 main loop: 4 consecutive elements = 4 consecutive variables
    for (int i = (int)(blockIdx.x * blockDim.x + threadIdx.x); i < n4; i += gsz) {
        float4 xv = ((const float4*)in)[i];
        int v0 = (i << 2) & (NV - 1);                 // multiple of 4, <= 60
        float4 ov, lv;
        eval_one(&S[(v0 + 0) * TSTRIDE], xv.x, ov.x, lv.x);
        eval_one(&S[(v0 + 1) * TSTRIDE], xv.y, ov.y, lv.y);
        eval_one(&S[(v0 + 2) * TSTRIDE], xv.z, ov.z, lv.z);
        eval_one(&S[(v0 + 3) * TSTRIDE], xv.w, ov.w, lv.w);
        ((float4*)out)[i] = ov;
        ((float4*)lad)[i] = lv;
    }
    // scalar tail (N not a multiple of 4)
    for (int t = (n4 << 2) + (int)(blockIdx.x * blockDim.x + threadIdx.x); t < N; t += gsz) {
        float x = in[t];
        int v = t & (NV - 1);
        float o, l;
        eval_one(&S[v * TSTRIDE], x, o, l);
        out[t] = o;
        lad[t] = l;
    }
}

// ---------------------------------------------------------------------------
extern "C" void kernel_launch(void* const* d_in, const int* in_sizes, int n_in,
                              void* d_out, int out_size, void* d_ws, size_t ws_size,
                              hipStream_t stream) {
    const float* in = (const float*)d_in[0];
    const float* uw = (const float*)d_in[1];
    const float* uh = (const float*)d_in[2];
    const float* ud = (const float*)d_in[3];
    float* ws  = (float*)d_ws;
    int    N   = in_sizes[0];
    float* out = (float*)d_out;
    float* lad = out + N;

    // Phase 1: one wave32 per variable (latency ~O(100) cycles, not ~us)
    rqs_param_kernel<<<NV, 32, 0, stream>>>(uw, uh, ud, ws);

    // Phase 2
    int n4 = N >> 2;
    int blocks = 1024;                                // ~4 float4 iters/thread at B*V=4M
    int needed = (n4 + 255) / 256;
    if (needed < blocks) blocks = needed;
    if (blocks < 1) blocks = 1;
    rqs_eval_kernel<<<blocks, 256, 0, stream>>>(in, ws, out, lad, n4, N);
}